// SymmetricTransitionDownBlockPaperv3_20899310862387
// MI455X (gfx1250) — compile-verified
//
#include <hip/hip_runtime.h>
#include <hip/hip_bf16.h>

#define EPSV 1e-5f

typedef __attribute__((ext_vector_type(2))) float v2f;
typedef __attribute__((ext_vector_type(8))) float v8f;

// ---------------------------------------------------------------------------
// Kernel 1: y[N,128] = relu(bn(x[N,64] @ Wl[64,128]))  via f32 WMMA 16x16x4
// One wave -> one 16x16 output tile. 8 waves/block -> 16 rows x 128 cols.
// ---------------------------------------------------------------------------
__global__ __launch_bounds__(256) void k_linear2(
    const float* __restrict__ x,  const float* __restrict__ Wl,
    const float* __restrict__ gl, const float* __restrict__ bl,
    const float* __restrict__ ml, const float* __restrict__ vl,
    float* __restrict__ y, int N)
{
    const int lane = threadIdx.x & 31;
    const int wave = threadIdx.x >> 5;
    const int lo   = lane & 15;
    const int hi   = lane >> 4;            // 0 or 1
    const int row0 = blockIdx.x * 16;
    const int col0 = wave * 16;

    const int r = row0 + lo;
    const float* xrow = x  + (size_t)r * 64 + 2 * hi;           // A: row lo, K base 2*hi
    const float* wcol = Wl + (size_t)(2 * hi) * 128 + col0 + lo;// B: K base 2*hi, col lo

    v8f acc = {};
    #pragma unroll
    for (int k0 = 0; k0 < 64; k0 += 4) {
        v2f a; a.x = xrow[k0];            a.y = xrow[k0 + 1];
        v2f b; b.x = wcol[(size_t)k0*128]; b.y = wcol[(size_t)k0*128 + 128];
        acc = __builtin_amdgcn_wmma_f32_16x16x4_f32(
            false, a, false, b, (short)0, acc, false, false);
    }

    const int c  = col0 + lo;
    const float sc = gl[c] * rsqrtf(vl[c] + EPSV);
    const float sh = bl[c] - ml[c] * sc;
    #pragma unroll
    for (int j = 0; j < 8; ++j) {
        const int rr = row0 + j + 8 * hi;   // D vgpr j -> row j (+8 hi half)
        y[(size_t)rr * 128 + c] = fmaxf(acc[j] * sc + sh, 0.0f);
    }
}

// ---------------------------------------------------------------------------
// Kernel 2: per-center fused attention. One wave per center m.
//   h[16,64] = relu(bn([rel | x[knn]] @ W1))   (67 K-dims, padded to 68)
//   s = h @ W2 ; prob = softmax_K(s)
//   out[m,:] = sum_k prob[k] * y[knn[m,k],:]
// ---------------------------------------------------------------------------
__global__ __launch_bounds__(256) void k_attn(
    const float* __restrict__ p,   const float* __restrict__ x,
    const float* __restrict__ W1,  const float* __restrict__ g1,
    const float* __restrict__ b1,  const float* __restrict__ m1,
    const float* __restrict__ v1,  const float* __restrict__ W2,
    const int*   __restrict__ idx, const int* __restrict__ knn,
    const float* __restrict__ y,
    float* __restrict__ out_np, float* __restrict__ out_feat, int M)
{
    const int lane = threadIdx.x & 31;
    const int wave = threadIdx.x >> 5;
    const int m    = blockIdx.x * 8 + wave;
    if (m >= M) return;                    // wave-uniform

    const int lo = lane & 15;
    const int hi = lane >> 4;

    const int g    = knn[m * 16 + lo];     // neighbor index for row `lo`
    const int cidx = idx[m];
    const float* xg = x + (size_t)g * 64;

    if (lane < 3) out_np[m * 3 + lane] = p[cidx * 3 + lane];

    v8f accs[4] = {{}, {}, {}, {}};

    #pragma unroll
    for (int chunk = 0; chunk <= 16; ++chunk) {
        const int kk0 = chunk * 4 + 2 * hi;      // first of the two K indices
        // ---- A fragment: x_knn[row=lo][kk0], x_knn[row=lo][kk0+1]
        v2f a;
        if (chunk == 0) {
            if (hi == 0) {  // kk = 0,1 : rel xyz
                a.x = p[g * 3 + 0] - p[cidx * 3 + 0];
                a.y = p[g * 3 + 1] - p[cidx * 3 + 1];
            } else {        // kk = 2 (rel z), kk = 3 (x[...,0])
                a.x = p[g * 3 + 2] - p[cidx * 3 + 2];
                a.y = xg[0];
            }
        } else if (chunk == 16) {
            if (hi == 0) { a.x = xg[61]; a.y = xg[62]; }   // kk = 64,65
            else         { a.x = xg[63]; a.y = 0.0f;   }   // kk = 66, 67=pad
        } else {
            a.x = xg[kk0 - 3]; a.y = xg[kk0 - 2];          // kk in [4,63]
        }
        const bool padB = (chunk == 16) && (hi == 1);      // kk0+1 == 67 -> zero row
        // ---- 4 column tiles of W1 [67,64]
        #pragma unroll
        for (int t = 0; t < 4; ++t) {
            const float* wbt = W1 + (size_t)kk0 * 64 + t * 16 + lo;
            v2f b; b.x = wbt[0]; b.y = padB ? 0.0f : wbt[64];
            accs[t] = __builtin_amdgcn_wmma_f32_16x16x4_f32(
                false, a, false, b, (short)0, accs[t], false, false);
        }
    }

    // ---- BN + ReLU on h, then per-lane partial of s = h @ W2
    float ps[8];
    #pragma unroll
    for (int j = 0; j < 8; ++j) ps[j] = 0.0f;
    #pragma unroll
    for (int t = 0; t < 4; ++t) {
        const int c = t * 16 + lo;
        const float sc = g1[c] * rsqrtf(v1[c] + EPSV);
        const float sh = b1[c] - m1[c] * sc;
        const float w2 = W2[c];
        #pragma unroll
        for (int j = 0; j < 8; ++j) {
            const float hv = fmaxf(accs[t][j] * sc + sh, 0.0f);
            ps[j] += hv * w2;
        }
    }
    // reduce over the 16 lanes of each half -> ps[j] = s[row j + 8*hi]
    #pragma unroll
    for (int j = 0; j < 8; ++j) {
        ps[j] += __shfl_xor(ps[j], 1);
        ps[j] += __shfl_xor(ps[j], 2);
        ps[j] += __shfl_xor(ps[j], 4);
        ps[j] += __shfl_xor(ps[j], 8);
    }
    // softmax over the 16 rows (b2 cancels)
    float mx = ps[0];
    #pragma unroll
    for (int j = 1; j < 8; ++j) mx = fmaxf(mx, ps[j]);
    mx = fmaxf(mx, __shfl_xor(mx, 16));
    float e[8]; float sum = 0.0f;
    #pragma unroll
    for (int j = 0; j < 8; ++j) { e[j] = __expf(ps[j] - mx); sum += e[j]; }
    sum += __shfl_xor(sum, 16);
    const float inv = 1.0f / sum;

    // ---- weighted gather-sum over y: each lane owns 4 of 128 channels
    float o0 = 0.f, o1 = 0.f, o2 = 0.f, o3 = 0.f;
    #pragma unroll
    for (int r = 0; r < 16; ++r) {
        const float pr = __shfl(e[r & 7], (r < 8) ? 0 : 16) * inv;
        const int   gr = __shfl(g, r);
        const float4 yv = *(const float4*)(y + (size_t)gr * 128 + lane * 4);
        o0 += pr * yv.x; o1 += pr * yv.y; o2 += pr * yv.z; o3 += pr * yv.w;
    }
    float4 o; o.x = o0; o.y = o1; o.z = o2; o.w = o3;
    *(float4*)(out_feat + (size_t)m * 128 + lane * 4) = o;
}

// ---------------------------------------------------------------------------
extern "C" void kernel_launch(void* const* d_in, const int* in_sizes, int n_in,
                              void* d_out, int out_size, void* d_ws, size_t ws_size,
                              hipStream_t stream) {
    const float* p  = (const float*)d_in[0];
    const float* x  = (const float*)d_in[1];
    const float* W1 = (const float*)d_in[2];
    const float* g1 = (const float*)d_in[3];
    const float* b1 = (const float*)d_in[4];
    const float* m1 = (const float*)d_in[5];
    const float* v1 = (const float*)d_in[6];
    const float* W2 = (const float*)d_in[7];
    // d_in[8] = b2 (cancels in softmax)
    const float* Wl = (const float*)d_in[9];
    const float* gl = (const float*)d_in[10];
    const float* bl = (const float*)d_in[11];
    const float* ml = (const float*)d_in[12];
    const float* vl = (const float*)d_in[13];
    const int* idx  = (const int*)d_in[14];
    const int* knn  = (const int*)d_in[15];

    const int N = in_sizes[1] / 64;   // x is [N,64]
    const int M = in_sizes[14];       // idx is [M]

    float* y        = (float*)d_ws;              // [N,128] scratch, L2-resident
    float* out_np   = (float*)d_out;             // [M,3]
    float* out_feat = (float*)d_out + (size_t)M * 3; // [M,128]

    // Kernel 1: y = relu(bn(x @ Wl))
    k_linear2<<<N / 16, 256, 0, stream>>>(x, Wl, gl, bl, ml, vl, y, N);

    // Kernel 2: fused shrinker + softmax + weighted gather
    k_attn<<<(M + 7) / 8, 256, 0, stream>>>(p, x, W1, g1, b1, m1, v1, W2,
                                            idx, knn, y, out_np, out_feat, M);
}